// Focal_Loss_10393820856644
// MI455X (gfx1250) — compile-verified
//
#include <hip/hip_runtime.h>

typedef __attribute__((ext_vector_type(2))) float v2f;
typedef __attribute__((ext_vector_type(8))) float v8f;

#define FL_EPS 1e-7f
#define LN2F   0.6931471805599453f

// Swap data between the two 16-lane halves of a wave32 (== __shfl_xor(x,16,32))
// entirely in the VALU: v_permlanex16_b32 with identity lane selects.
// Avoids ds_bpermute_b32 + s_wait_dscnt in the hot loop.
__device__ __forceinline__ float xhalf_swap(float x) {
    int v = __builtin_amdgcn_permlanex16(__float_as_int(x), __float_as_int(x),
                                         0x76543210, 0xFEDCBA98, true, false);
    return __int_as_float(v);
}

// One wave processes a 16-row x 16-class tile, laid out as the A-matrix of
// V_WMMA_F32_16X16X4_F32:
//   lane L in [0,16):  row L,    K = c*4 + {0,1} per chunk c (VGPR0/1 of A)
//   lane L in [16,32): row L-16, K = c*4 + {2,3} per chunk c
// Row-sum of exp() runs on the matrix pipe as D += A_c * ones(4x16),
// co-executing with the VALU exp/log work.
__global__ __launch_bounds__(256) void focal_main(
    const float* __restrict__ pred,
    const int*   __restrict__ target,
    const float* __restrict__ alpha,
    float* __restrict__ partials,
    int numTiles)
{
    const int lane          = threadIdx.x & 31;
    const int waveId        = threadIdx.x >> 5;
    const int wavesPerBlock = blockDim.x >> 5;
    const int gWave         = blockIdx.x * wavesPerBlock + waveId;
    const int nWaves        = gridDim.x * wavesPerBlock;
    const int row16         = lane & 15;   // row within tile
    const int hf            = lane >> 4;   // which K-pair of each 4-chunk

    const v2f* __restrict__ pred2 = (const v2f*)pred;

    float acc = 0.0f;

    #pragma unroll 2
    for (int tile = gWave; tile < numTiles; tile += nWaves) {
        const int    r    = tile * 16 + row16;
        const size_t base = (size_t)r * 8 + (size_t)hf;  // float2 units, 8 per row

        // 4 x global_load_b64; across the wave these cover the tile's
        // contiguous 1KiB of pred.
        v2f f0 = pred2[base + 0];
        v2f f1 = pred2[base + 2];
        v2f f2 = pred2[base + 4];
        v2f f3 = pred2[base + 6];
        int t  = target[r];

        // Stable-softmax row max: local max of 8, then half-swap via permlanex16.
        float m = fmaxf(fmaxf(fmaxf(f0.x, f0.y), fmaxf(f1.x, f1.y)),
                        fmaxf(fmaxf(f2.x, f2.y), fmaxf(f3.x, f3.y)));
        m = fmaxf(m, xhalf_swap(m));

        v2f e0, e1, e2, e3;
        e0.x = __expf(f0.x - m); e0.y = __expf(f0.y - m);
        e1.x = __expf(f1.x - m); e1.y = __expf(f1.y - m);
        e2.x = __expf(f2.x - m); e2.y = __expf(f2.y - m);
        e3.x = __expf(f3.x - m); e3.y = __expf(f3.y - m);

        // Numerator exp at the target column. This lane owns columns
        // c*4 + hf*2 + {0,1}; the partner lane owns the rest.
        const int c0 = hf * 2;
        float et = 0.0f;
        et = (t == c0 + 0)  ? e0.x : et;
        et = (t == c0 + 1)  ? e0.y : et;
        et = (t == c0 + 4)  ? e1.x : et;
        et = (t == c0 + 5)  ? e1.y : et;
        et = (t == c0 + 8)  ? e2.x : et;
        et = (t == c0 + 9)  ? e2.y : et;
        et = (t == c0 + 12) ? e3.x : et;
        et = (t == c0 + 13) ? e3.y : et;
        et += xhalf_swap(et);

        // Row sums on the matrix pipe: D[m][n] = sum_k exp[m][k].
        v2f ones; ones.x = 1.0f; ones.y = 1.0f;
        v8f d = {};
        d = __builtin_amdgcn_wmma_f32_16x16x4_f32(false, e0, false, ones, (short)0, d, false, false);
        d = __builtin_amdgcn_wmma_f32_16x16x4_f32(false, e1, false, ones, (short)0, d, false, false);
        d = __builtin_amdgcn_wmma_f32_16x16x4_f32(false, e2, false, ones, (short)0, d, false, false);
        d = __builtin_amdgcn_wmma_f32_16x16x4_f32(false, e3, false, ones, (short)0, d, false, false);

        // D layout: lanes 0-15 hold rows 0-7 in d[0..7]; lanes 16-31 rows 8-15.
        // Select d[row & 7] with a cndmask tree, then fetch across the half
        // boundary if this lane's row lives in the other half.
        const int r8 = row16 & 7;
        float a0 = (r8 & 1) ? d[1] : d[0];
        float a1 = (r8 & 1) ? d[3] : d[2];
        float a2 = (r8 & 1) ? d[5] : d[4];
        float a3 = (r8 & 1) ? d[7] : d[6];
        float b0 = (r8 & 2) ? a1 : a0;
        float b1 = (r8 & 2) ? a3 : a2;
        float s  = (r8 & 4) ? b1 : b0;
        float sOther  = xhalf_swap(s);
        bool  haveRow = (hf == 0) ? (row16 < 8) : (row16 >= 8);
        float rowsum  = haveRow ? s : sOther;

        // rowsum >= exp(max-max) == 1, so a raw v_rcp_f32 is safe (no div fixup).
        float p    = et * __builtin_amdgcn_rcpf(rowsum);
        // p + eps in [1e-7, ~2]: always normal, raw v_log_f32 (log2) is safe.
        float loss = -LN2F * __builtin_amdgcn_logf(p + FL_EPS);
        float om   = 1.0f - p;
        float w    = om * om * alpha[t];   // gamma == 2
        acc += (hf == 0) ? (w * loss) : 0.0f;  // count each row once
    }

    // Deterministic block reduction: wave shuffle tree + LDS across waves.
    for (int off = 16; off > 0; off >>= 1)
        acc += __shfl_down(acc, off, 32);

    __shared__ float sred[32];
    if (lane == 0) sred[waveId] = acc;
    __syncthreads();
    if (waveId == 0) {
        float v = (lane < wavesPerBlock) ? sred[lane] : 0.0f;
        for (int off = 16; off > 0; off >>= 1)
            v += __shfl_down(v, off, 32);
        if (lane == 0) partials[blockIdx.x] = v;
    }
}

__global__ __launch_bounds__(256) void focal_reduce(
    const float* __restrict__ partials, int n,
    float* __restrict__ out, float invN)
{
    float s = 0.0f;
    for (int i = threadIdx.x; i < n; i += blockDim.x)
        s += partials[i];

    const int lane = threadIdx.x & 31;
    const int w    = threadIdx.x >> 5;
    for (int off = 16; off > 0; off >>= 1)
        s += __shfl_down(s, off, 32);

    __shared__ float sm[32];
    if (lane == 0) sm[w] = s;
    __syncthreads();
    if (w == 0) {
        float v = (lane < (int)(blockDim.x >> 5)) ? sm[lane] : 0.0f;
        for (int off = 16; off > 0; off >>= 1)
            v += __shfl_down(v, off, 32);
        if (lane == 0) out[0] = v * invN;
    }
}

extern "C" void kernel_launch(void* const* d_in, const int* in_sizes, int n_in,
                              void* d_out, int out_size, void* d_ws, size_t ws_size,
                              hipStream_t stream) {
    const float* pred   = (const float*)d_in[0];  // [N, 16] f32
    const int*   target = (const int*)  d_in[1];  // [N] int
    const float* alpha  = (const float*)d_in[2];  // [16] f32
    float*       out    = (float*)d_out;          // scalar
    float*       parts  = (float*)d_ws;

    const int N        = in_sizes[1];   // number of rows (target count)
    const int numTiles = N / 16;        // N = 4M is divisible by 16

    const int BLOCKS = 4096;            // 32768 wave32s; 8 tiles per wave
    const int TPB    = 256;             // 8 waves / block

    focal_main<<<BLOCKS, TPB, 0, stream>>>(pred, target, alpha, parts, numTiles);
    focal_reduce<<<1, TPB, 0, stream>>>(parts, BLOCKS, out, 1.0f / (float)N);
}